// pros_71751723647384
// MI455X (gfx1250) — compile-verified
//
#include <hip/hip_runtime.h>
#include <stdint.h>

typedef uint32_t u32;

#define TPB   256
#define TILE  4096          // elements per block (TPB * 16)
#define EPT   16            // elements per thread in scatter pass

// ---------------------------------------------------------------------------
// Per-element classification. Emits (segment, value) for up to 4 pieces.
// Segment = bin*8 + piece, piece order matches the torch.cat order exactly.
// Float op order matches the jax reference bit-for-bit intent:
//   p1: (d-bl)+bl   p2: S   p3: (S+bl)-bl   p4: d
//   p5: (d+br)-br   p6: E   p7: (E-br)+br   p8: S
// Thresholds e-0.01 / s+0.01 computed in double then cast to f32 (numpy rules).
// ---------------------------------------------------------------------------
template <class F>
__device__ __forceinline__ void classify(float draw, float lab,
                                         float4 tlv, float4 trv, F&& emit) {
  float d = draw < 0.0f ? 0.0f : (draw > 1.0f ? 1.0f : draw);
  int b = (int)(d * 4.0f);              // *4 is exact (power of two)
  if (b > 3) return;                    // d == 1.0 -> no bin
  float S = 0.25f * (float)b;           // exact
  float E = S + 0.25f;                  // exact
  if (!(d > S && d < E)) return;        // open interval (bin edges excluded)

  float bl = (b == 0) ? tlv.x : (b == 1) ? tlv.y : (b == 2) ? tlv.z : tlv.w;
  float br = (b == 0) ? trv.x : (b == 1) ? trv.y : (b == 2) ? trv.z : trv.w;

  const float TOLt = (b == 0) ? (float)(0.25 * 1 - 0.01)
                   : (b == 1) ? (float)(0.25 * 2 - 0.01)
                   : (b == 2) ? (float)(0.25 * 3 - 0.01)
                              : (float)(0.25 * 4 - 0.01);
  const float TORt = (b == 0) ? (float)(0.25 * 0 + 0.01)
                   : (b == 1) ? (float)(0.25 * 1 + 0.01)
                   : (b == 2) ? (float)(0.25 * 2 + 0.01)
                              : (float)(0.25 * 3 + 0.01);

  bool tol = lab < TOLt;
  bool tor = lab > TORt;
  bool mlp = ((d - S) - bl) > 0.0f;     // ml = (d - s) - tl, f32 order
  bool mrp = ((d - E) + br) > 0.0f;     // mr = (d - e) + tr, f32 order
  int base = b * 8;

  if (tol) emit(base + (mlp ? 0 : 1), mlp ? ((d - bl) + bl) : S);
  if (tor) emit(base + (mlp ? 3 : 2), mlp ? d : ((S + bl) - bl));
  if (tor) emit(base + (mrp ? 5 : 4), mrp ? E : ((d + br) - br));
  if (tol) emit(base + (mrp ? 6 : 7), mrp ? ((E - br) + br) : S);
}

// ---------------------------------------------------------------------------
// Pass 1: per-block per-segment counts. Counters live in owned LDS columns
// (u16 pairs packed in u32: segs (2k,2k+1) share word k). Tree-reduced, then
// written seg-major: counts[seg*numBlocks + blk].
// ---------------------------------------------------------------------------
__global__ void __launch_bounds__(TPB)
k_count(const float* __restrict__ data, const float* __restrict__ label,
        const float* __restrict__ tl, const float* __restrict__ tr,
        u32* __restrict__ counts, int numBlocks, int N) {
  __shared__ u32 sScan[16 * TPB];
  const int tid = threadIdx.x;
  const int blk = blockIdx.x;
  const long long tileStart = (long long)blk * TILE;

  float4 tlv = *(const float4*)tl;
  float4 trv = *(const float4*)tr;

#pragma unroll
  for (int s = 0; s < 16; ++s) sScan[s * TPB + tid] = 0u;   // own column

  auto emitCnt = [&](int seg, float) {
    sScan[(seg >> 1) * TPB + tid] += 1u << ((seg & 1) * 16);
  };

  if (tileStart + TILE <= (long long)N) {
    const float4* d4 = (const float4*)(data + tileStart);
    const float4* l4 = (const float4*)(label + tileStart);
#pragma unroll
    for (int r = 0; r < TILE / (TPB * 4); ++r) {
      float4 dv = d4[r * TPB + tid];
      float4 lv = l4[r * TPB + tid];
      classify(dv.x, lv.x, tlv, trv, emitCnt);
      classify(dv.y, lv.y, tlv, trv, emitCnt);
      classify(dv.z, lv.z, tlv, trv, emitCnt);
      classify(dv.w, lv.w, tlv, trv, emitCnt);
    }
  } else {
    for (int r = 0; r < TILE / TPB; ++r) {
      long long e = tileStart + r * TPB + tid;
      if (e < N) classify(data[e], label[e], tlv, trv, emitCnt);
    }
  }
  __syncthreads();

#pragma unroll
  for (int off = TPB / 2; off >= 1; off >>= 1) {
    if (tid < off) {
#pragma unroll
      for (int s = 0; s < 16; ++s)
        sScan[s * TPB + tid] += sScan[s * TPB + tid + off];
    }
    __syncthreads();
  }
  if (tid < 16) {
    u32 w = sScan[tid * TPB];
    counts[(size_t)(2 * tid + 0) * numBlocks + blk] = w & 0xFFFFu;
    counts[(size_t)(2 * tid + 1) * numBlocks + blk] = w >> 16;
  }
}

// ---------------------------------------------------------------------------
// Pass 2a: per-segment exclusive scan over block counts (in place).
// One block per segment. Thread t owns a contiguous chunk of blocks.
// ---------------------------------------------------------------------------
__global__ void __launch_bounds__(TPB)
k_scanblocks(u32* __restrict__ counts, u32* __restrict__ totals, int numBlocks) {
  __shared__ u32 sm[TPB];
  const int seg = blockIdx.x;
  const int tid = threadIdx.x;
  const int C = (numBlocks + TPB - 1) / TPB;
  u32* row = counts + (size_t)seg * numBlocks;

  u32 sum = 0;
  for (int j = 0; j < C; ++j) {
    int idx = tid * C + j;
    if (idx < numBlocks) sum += row[idx];
  }
  sm[tid] = sum;
  __syncthreads();
  for (int off = 1; off < TPB; off <<= 1) {
    u32 t = (tid >= off) ? sm[tid - off] : 0u;
    __syncthreads();
    sm[tid] += t;
    __syncthreads();
  }
  if (tid == TPB - 1) totals[seg] = sm[tid];
  u32 run = sm[tid] - sum;              // exclusive base for this thread
  for (int j = 0; j < C; ++j) {
    int idx = tid * C + j;
    if (idx < numBlocks) {
      u32 c = row[idx];
      row[idx] = run;
      run += c;
    }
  }
}

// Pass 2b: segment bases (exclusive scan over 32 totals) + total M.
__global__ void k_segbase(const u32* __restrict__ totals, u32* __restrict__ segBase) {
  if (threadIdx.x == 0) {
    u32 run = 0;
    for (int s = 0; s < 32; ++s) { segBase[s] = run; run += totals[s]; }
    segBase[32] = run;   // M
  }
}

// ---------------------------------------------------------------------------
// Pass 3: stable scatter. Stage data+label tiles into LDS with CDNA5 async
// loads (ASYNCcnt path), classify from registers, do a 256-thread x 32-seg
// packed Kogge-Stone prefix in LDS, then write outputs with NON-TEMPORAL
// stores (write-once stream; keeps the twice-read 64MB input resident in the
// 192MB L2). Thread t owns elements [t*16, t*16+16): block order + thread
// order + element order == global index order => stable compaction.
// ---------------------------------------------------------------------------
__global__ void __launch_bounds__(TPB)
k_scatter(const float* __restrict__ data, const float* __restrict__ label,
          const float* __restrict__ tl, const float* __restrict__ tr,
          const u32* __restrict__ blockOff,    // exclusive offsets, seg-major
          const u32* __restrict__ segBase,
          float* __restrict__ out, int numBlocks, int N) {
  __shared__ float sData[TILE];
  __shared__ float sLab[TILE];
  __shared__ u32   sScan[16 * TPB];
  __shared__ u32   sBase[32];
  __shared__ u32   sM;

  const int tid = threadIdx.x;
  const int blk = blockIdx.x;
  const long long tileStart = (long long)blk * TILE;
  const bool full = (tileStart + TILE <= (long long)N);

  if (full) {
    // CDNA5 async copy global -> LDS, 16B per lane per op (ASYNCcnt path).
    u32 ldsD = (u32)(uintptr_t)&sData[0];
    u32 ldsL = (u32)(uintptr_t)&sLab[0];
    const char* gD = (const char*)(data + tileStart);
    const char* gL = (const char*)(label + tileStart);
#pragma unroll
    for (int r = 0; r < (TILE * 4) / (TPB * 16); ++r) {   // 4 rounds each
      u32 byteOff = (u32)(r * TPB + tid) * 16u;
      uint64_t gaD = (uint64_t)(uintptr_t)(gD + byteOff);
      uint64_t gaL = (uint64_t)(uintptr_t)(gL + byteOff);
      asm volatile("global_load_async_to_lds_b128 %0, %1, off"
                   :: "v"(ldsD + byteOff), "v"(gaD) : "memory");
      asm volatile("global_load_async_to_lds_b128 %0, %1, off"
                   :: "v"(ldsL + byteOff), "v"(gaL) : "memory");
    }
  } else {
    for (int r = 0; r < TILE / TPB; ++r) {
      int e = r * TPB + tid;
      long long g = tileStart + e;
      sData[e] = (g < (long long)N) ? data[g] : 2.0f;   // 2.0 -> clamps to 1 -> no bin
      sLab[e]  = (g < (long long)N) ? label[g] : 0.0f;
    }
  }

  if (tid < 32)
    sBase[tid] = segBase[tid] + blockOff[(size_t)tid * numBlocks + blk];
  if (tid == 0) sM = segBase[32];

#pragma unroll
  for (int s = 0; s < 16; ++s) sScan[s * TPB + tid] = 0u;  // own column

  if (full) asm volatile("s_wait_asynccnt 0" ::: "memory");
  __syncthreads();

  float4 tlv = *(const float4*)tl;
  float4 trv = *(const float4*)tr;

  // My 16 contiguous elements -> registers (ds_load_b128 x4 each).
  float dv[EPT], lv[EPT];
#pragma unroll
  for (int q = 0; q < EPT / 4; ++q) {
    float4 a = ((const float4*)sData)[tid * (EPT / 4) + q];
    float4 b = ((const float4*)sLab)[tid * (EPT / 4) + q];
    dv[q * 4 + 0] = a.x; dv[q * 4 + 1] = a.y; dv[q * 4 + 2] = a.z; dv[q * 4 + 3] = a.w;
    lv[q * 4 + 0] = b.x; lv[q * 4 + 1] = b.y; lv[q * 4 + 2] = b.z; lv[q * 4 + 3] = b.w;
  }

  auto emitCnt = [&](int seg, float) {
    sScan[(seg >> 1) * TPB + tid] += 1u << ((seg & 1) * 16);
  };
#pragma unroll
  for (int j = 0; j < EPT; ++j) classify(dv[j], lv[j], tlv, trv, emitCnt);
  __syncthreads();

  // Kogge-Stone inclusive scan along tid, 16 packed words per thread.
  for (int off = 1; off < TPB; off <<= 1) {
    u32 t[16];
#pragma unroll
    for (int s = 0; s < 16; ++s)
      t[s] = (tid >= off) ? sScan[s * TPB + tid - off] : 0u;
    __syncthreads();
#pragma unroll
    for (int s = 0; s < 16; ++s) sScan[s * TPB + tid] += t[s];
    __syncthreads();
  }

  // Replace own column with thread-exclusive running counters.
  u32 excl[16];
#pragma unroll
  for (int s = 0; s < 16; ++s)
    excl[s] = (tid > 0) ? sScan[s * TPB + tid - 1] : 0u;
  __syncthreads();
#pragma unroll
  for (int s = 0; s < 16; ++s) sScan[s * TPB + tid] = excl[s];
  // Column is thread-private from here on; no further barriers needed.

  const u32 M = sM;
  float curLab;
  auto emitWr = [&](int seg, float val) {
    int   w   = (seg >> 1) * TPB + tid;
    int   sh  = (seg & 1) * 16;
    u32   cur = sScan[w];
    sScan[w]  = cur + (1u << sh);
    u32 rank  = (cur >> sh) & 0xFFFFu;
    u32 idx   = sBase[seg] + rank;
    // Write-once streaming output: non-temporal (th:TH_STORE_NT) so the
    // ~230MB output stream does not evict the L2-resident inputs.
    __builtin_nontemporal_store(val,    &out[idx]);       // d-region
    __builtin_nontemporal_store(curLab, &out[M + idx]);   // label-region
  };
#pragma unroll
  for (int j = 0; j < EPT; ++j) {
    curLab = lv[j];
    classify(dv[j], lv[j], tlv, trv, emitWr);
  }
}

// ---------------------------------------------------------------------------
// Workspace layout (u32 units):
//   [0..32]   segBase (32 bases + M at [32])
//   [96..127] per-segment totals
//   [128 ..]  block counts / offsets, seg-major: 32 * numBlocks words
// ---------------------------------------------------------------------------
extern "C" void kernel_launch(void* const* d_in, const int* in_sizes, int n_in,
                              void* d_out, int out_size, void* d_ws, size_t ws_size,
                              hipStream_t stream) {
  const float* data  = (const float*)d_in[0];
  const float* label = (const float*)d_in[1];
  const float* tl    = (const float*)d_in[2];
  const float* tr    = (const float*)d_in[3];
  const int N = in_sizes[0];
  const int numBlocks = (N + TILE - 1) / TILE;

  u32* wsU     = (u32*)d_ws;
  u32* segBase = wsU;
  u32* totals  = wsU + 96;
  u32* counts  = wsU + 128;

  k_count<<<numBlocks, TPB, 0, stream>>>(data, label, tl, tr, counts, numBlocks, N);
  k_scanblocks<<<32, TPB, 0, stream>>>(counts, totals, numBlocks);
  k_segbase<<<1, 32, 0, stream>>>(totals, segBase);
  k_scatter<<<numBlocks, TPB, 0, stream>>>(data, label, tl, tr, counts, segBase,
                                           (float*)d_out, numBlocks, N);
}